// NonMaxSuppression_13786845020725
// MI455X (gfx1250) — compile-verified
//
#include <hip/hip_runtime.h>
#include <stdint.h>

// Problem constants (reference: repeatability (16,1,1536,1536) fp32)
#define NB        16
#define H         1536
#define W         1536
#define BORDER    10
#define THR       0.6f
#define VALID     (H - 2 * BORDER)                        // 1516 valid rows/cols
#define ROWS_PER_BLK 8                                    // one row per wave, 8 waves/block
#define CHUNKS    ((VALID + ROWS_PER_BLK - 1) / ROWS_PER_BLK)  // 190 row-chunks per image
#define R_TOT     (NB * VALID)                            // 24256 valid rows total
#define SLAB_ROWS (ROWS_PER_BLK + 2)                      // 10 rows incl. halo
#define SLAB_FLOATS (SLAB_ROWS * W)                       // 15360 floats
#define SLAB_BYTES  (SLAB_FLOATS * 4)                     // 61440 bytes of LDS
#define B128_CNT    (SLAB_BYTES / 16)                     // 3840 b128 async transfers

#define BLK_THREADS 256

// wave32 ballot
#if defined(__has_builtin)
#  if __has_builtin(__builtin_amdgcn_ballot_w32)
#    define BALLOT32(p) __builtin_amdgcn_ballot_w32(p)
#  endif
#endif
#ifndef BALLOT32
#  define BALLOT32(p) ((unsigned)__ballot(p))
#endif

// ---------------------------------------------------------------------------
// CDNA5 async global->LDS staging of one contiguous 10-row slab.
// global_load_async_to_lds_b128: vdst = LDS byte address VGPR, vaddr = 32-bit
// per-lane byte offset, saddr = uniform 64-bit base (GVS mode). Tracked by
// ASYNCcnt; completion enforced with s_wait_asynccnt.
// ---------------------------------------------------------------------------
__device__ __forceinline__ void async_tile_load(const float* gbase, unsigned lds_base) {
  for (int j = threadIdx.x; j < B128_CNT; j += BLK_THREADS) {
    unsigned off = (unsigned)j * 16u;
    unsigned lds_addr = lds_base + off;
    asm volatile("global_load_async_to_lds_b128 %0, %1, %2"
                 :: "v"(lds_addr), "v"(off), "s"(gbase)
                 : "memory");
  }
  asm volatile("s_wait_asynccnt 0x0" ::: "memory");
}

// 3x3 NMS predicate read from LDS slab. srow = row within slab (1..8).
// Tie semantics match reference: v == max(3x3) <=> v >= max(8 neighbors).
__device__ __forceinline__ bool nms_pred(const float* tile, int srow, int x) {
  const float* r0 = tile + (srow - 1) * W;
  const float* r1 = tile + srow * W;
  const float* r2 = tile + (srow + 1) * W;
  float v = r1[x];
  float m = r0[x - 1];
  m = fmaxf(m, r0[x]);
  m = fmaxf(m, r0[x + 1]);
  m = fmaxf(m, r1[x - 1]);
  m = fmaxf(m, r1[x + 1]);
  m = fmaxf(m, r2[x - 1]);
  m = fmaxf(m, r2[x]);
  m = fmaxf(m, r2[x + 1]);
  return (v >= THR) && (v >= m);
}

// ---------------------------------------------------------------------------
// Pass 1: per-row survivor counts. Block = 8 waves, wave w owns row y0+w.
// ---------------------------------------------------------------------------
__global__ void __launch_bounds__(BLK_THREADS)
nms_count(const float* __restrict__ rep, unsigned* __restrict__ counts) {
  __shared__ float tile[SLAB_FLOATS];
  const int blk   = blockIdx.x;
  const int b     = blk / CHUNKS;
  const int chunk = blk % CHUNKS;
  const int y0    = BORDER + chunk * ROWS_PER_BLK;

  const float* gbase = rep + ((size_t)b * H + (size_t)(y0 - 1)) * W;
  const unsigned tbase = (unsigned)(unsigned long long)(uintptr_t)&tile[0];
  async_tile_load(gbase, tbase);
  __syncthreads();

  const int wave = threadIdx.x >> 5;
  const int lane = threadIdx.x & 31;
  const int y = y0 + wave;
  if (y <= H - BORDER - 1) {
    unsigned cnt = 0;
    for (int xc = BORDER; xc < W - BORDER; xc += 32) {
      const int x = xc + lane;
      bool p = false;
      if (x < W - BORDER) p = nms_pred(tile, wave + 1, x);
      const unsigned mask = BALLOT32(p);
      cnt += (unsigned)__popc(mask);
    }
    if (lane == 0) counts[b * VALID + (y - BORDER)] = cnt;
  }
}

// ---------------------------------------------------------------------------
// Pass 2: exclusive prefix sum over R_TOT row counts (single 1024-thr block).
// ---------------------------------------------------------------------------
#define SCAN_T 1024
#define PER_T  ((R_TOT + SCAN_T - 1) / SCAN_T)   // 24

__global__ void __launch_bounds__(SCAN_T)
nms_scan(const unsigned* __restrict__ counts, unsigned* __restrict__ offsets) {
  __shared__ unsigned s[SCAN_T];
  const int t  = threadIdx.x;
  const int i0 = t * PER_T;

  unsigned loc[PER_T];
  unsigned sum = 0;
#pragma unroll
  for (int k = 0; k < PER_T; ++k) {
    const int i = i0 + k;
    const unsigned c = (i < R_TOT) ? counts[i] : 0u;
    loc[k] = sum;           // thread-local exclusive prefix
    sum += c;
  }
  s[t] = sum;
  __syncthreads();

  // Hillis-Steele inclusive scan over the 1024 per-thread sums
  for (int off = 1; off < SCAN_T; off <<= 1) {
    unsigned add = 0;
    if (t >= off) add = s[t - off];
    __syncthreads();
    if (t >= off) s[t] += add;
    __syncthreads();
  }

  const unsigned base = (t > 0) ? s[t - 1] : 0u;
#pragma unroll
  for (int k = 0; k < PER_T; ++k) {
    const int i = i0 + k;
    if (i < R_TOT) offsets[i] = base + loc[k];
  }
}

// ---------------------------------------------------------------------------
// Pass 3: ordered coordinate emission. out[0..K) = y, out[K..2K) = x.
// Ballot + lane-rank preserves ascending-x order inside each row; per-row
// exclusive offsets preserve the global (b, y, x) row-major order.
// ---------------------------------------------------------------------------
__global__ void __launch_bounds__(BLK_THREADS)
nms_emit(const float* __restrict__ rep, const unsigned* __restrict__ offsets,
         float* __restrict__ out, unsigned K) {
  __shared__ float tile[SLAB_FLOATS];
  const int blk   = blockIdx.x;
  const int b     = blk / CHUNKS;
  const int chunk = blk % CHUNKS;
  const int y0    = BORDER + chunk * ROWS_PER_BLK;

  const float* gbase = rep + ((size_t)b * H + (size_t)(y0 - 1)) * W;
  const unsigned tbase = (unsigned)(unsigned long long)(uintptr_t)&tile[0];
  async_tile_load(gbase, tbase);
  __syncthreads();

  const int wave = threadIdx.x >> 5;
  const int lane = threadIdx.x & 31;
  const int y = y0 + wave;
  if (y <= H - BORDER - 1) {
    unsigned base = offsets[b * VALID + (y - BORDER)];
    const unsigned ltmask = (1u << lane) - 1u;
    for (int xc = BORDER; xc < W - BORDER; xc += 32) {
      const int x = xc + lane;
      bool p = false;
      if (x < W - BORDER) p = nms_pred(tile, wave + 1, x);
      const unsigned mask = BALLOT32(p);
      if (p) {
        const unsigned rank = (unsigned)__popc(mask & ltmask);
        out[base + rank]     = (float)y;   // row 0 of [2,K]: y coords
        out[K + base + rank] = (float)x;   // row 1 of [2,K]: x coords
      }
      base += (unsigned)__popc(mask);
    }
  }
}

// ---------------------------------------------------------------------------
extern "C" void kernel_launch(void* const* d_in, const int* in_sizes, int n_in,
                              void* d_out, int out_size, void* d_ws, size_t ws_size,
                              hipStream_t stream) {
  (void)in_sizes; (void)n_in; (void)ws_size;
  const float* rep = (const float*)d_in[0];
  float* out = (float*)d_out;

  unsigned* counts  = (unsigned*)d_ws;          // R_TOT u32
  unsigned* offsets = counts + R_TOT;           // R_TOT u32

  const unsigned K = (unsigned)(out_size / 2);

  const dim3 blk(BLK_THREADS);
  const dim3 grid(NB * CHUNKS);                 // 3040 blocks

  hipLaunchKernelGGL(nms_count, grid, blk, 0, stream, rep, counts);
  hipLaunchKernelGGL(nms_scan, dim3(1), dim3(SCAN_T), 0, stream, counts, offsets);
  hipLaunchKernelGGL(nms_emit, grid, blk, 0, stream, rep, offsets, out, K);
}